// MeshMemory_83708912599405
// MI455X (gfx1250) — compile-verified
//
#include <hip/hip_runtime.h>
#include <hip/hip_bf16.h>
#include <math.h>

// MeshMemory fused update for MI455X (gfx1250, wave32, WMMA).
//
// Sizes: B=4, MAX_N=512, V=128, C=64, J=512, NUM_CLUTTER=5, BANK=4096.
// Outputs (flat, concatenated):
//   similarity       : 4*512*53248  = 109,051,904 floats
//   noise_similarity : 4*5*32768    =     655,360 floats
//   new_memory       : 64*128*512   =   4,194,304 floats
//   new_clutter_bank : 128*20480    =   2,621,440 floats

typedef float v2f __attribute__((ext_vector_type(2)));
typedef float v8f __attribute__((ext_vector_type(8)));

#define MAXN 512
#define VDIM 128
#define NCLS 64
#define NCLUT 5
#define BANK 4096
#define FEAT_ROWS (MAXN + NCLUT)          // 517
#define CJ (NCLS * MAXN)                  // 32768
#define CLUT_COLS (NCLUT * BANK)          // 20480
#define SIM_COLS (CJ + CLUT_COLS)         // 53248
#define NOISE_ROWS 20                     // B * NUM_CLUTTER

// ---------------------------------------------------------------------------
// similarity[b,i,n] = sum_v feats[b,i,v] * Wcat[v,n]
// Wcat[:, 0:32768]  = memory[c,v,j] with n = c*512 + j
// Wcat[:, 32768:..] = clutter_bank[v, n-32768]
// One workgroup (8 waves) = 512 M rows x 32 N cols of one batch.
// Each wave: 64M x 32N = 8 accumulators of v8f, K-loop step 4 (f32 WMMA).
// ---------------------------------------------------------------------------
__global__ __launch_bounds__(256) void sim_gemm_kernel(
    const float* __restrict__ feats,
    const float* __restrict__ memory,
    const float* __restrict__ clutter,
    float* __restrict__ out)
{
    const int b     = blockIdx.y;
    const int nBase = blockIdx.x * 32;
    const int wave  = threadIdx.x >> 5;
    const int lane  = threadIdx.x & 31;
    const int mBase = wave * 64;
    const int lrow  = lane & 15;
    const int khalf = (lane >> 4) << 1;        // 0 or 2

    const float* bptr;
    int bstride;
    if (nBase < CJ) {
        const int c  = nBase >> 9;
        const int j0 = nBase & 511;
        bptr    = memory + (size_t)c * (VDIM * MAXN) + j0;
        bstride = MAXN;                        // 512
    } else {
        bptr    = clutter + (nBase - CJ);
        bstride = CLUT_COLS;                   // 20480
    }

    const float* aptr = feats + (size_t)b * FEAT_ROWS * VDIM;

    v8f acc[4][2];
    #pragma unroll
    for (int ms = 0; ms < 4; ++ms)
        #pragma unroll
        for (int t = 0; t < 2; ++t)
            acc[ms][t] = v8f{0.f, 0.f, 0.f, 0.f, 0.f, 0.f, 0.f, 0.f};

    // Per-lane fixed A row pointers (row = mBase + ms*16 + lrow).
    const float* arow[4];
    #pragma unroll
    for (int ms = 0; ms < 4; ++ms)
        arow[ms] = aptr + (size_t)(mBase + ms * 16 + lrow) * VDIM;

    for (int k = 0; k < VDIM; k += 4) {
        // B fragments: lane holds B[k+khalf][col], B[k+khalf+1][col]
        v2f bf[2];
        #pragma unroll
        for (int t = 0; t < 2; ++t) {
            const float* p = bptr + (size_t)(k + khalf) * bstride + (t * 16 + lrow);
            bf[t].x = p[0];
            bf[t].y = p[bstride];
        }
        #pragma unroll
        for (int ms = 0; ms < 4; ++ms) {
            v2f af = *(const v2f*)(arow[ms] + k + khalf);
            #pragma unroll
            for (int t = 0; t < 2; ++t)
                acc[ms][t] = __builtin_amdgcn_wmma_f32_16x16x4_f32(
                    false, af, false, bf[t], (short)0, acc[ms][t], false, false);
        }
    }

    // C/D layout: VGPR r, lane l -> row = r + 8*(l>=16), col = l&15
    float* obase = out + (size_t)b * MAXN * SIM_COLS + nBase;
    const int rofs = (lane >> 4) << 3;         // 0 or 8
    #pragma unroll
    for (int ms = 0; ms < 4; ++ms)
        #pragma unroll
        for (int t = 0; t < 2; ++t)
            #pragma unroll
            for (int r = 0; r < 8; ++r) {
                const int m = mBase + ms * 16 + r + rofs;
                obase[(size_t)m * SIM_COLS + t * 16 + lrow] = acc[ms][t][r];
            }
}

// ---------------------------------------------------------------------------
// noise_similarity: 20 x 32768 GEMM vs memory (pad M to 32). One wave/block,
// each block covers a 16-wide column strip.
// ---------------------------------------------------------------------------
__global__ __launch_bounds__(32) void noise_gemm_kernel(
    const float* __restrict__ feats,
    const float* __restrict__ memory,
    float* __restrict__ out)
{
    const int nBase = blockIdx.x * 16;
    const int lane  = threadIdx.x;
    const int lrow  = lane & 15;
    const int khalf = (lane >> 4) << 1;

    const int c  = nBase >> 9;
    const int j0 = nBase & 511;
    const float* bptr = memory + (size_t)c * (VDIM * MAXN) + j0;

    v8f acc[2];
    acc[0] = v8f{0.f, 0.f, 0.f, 0.f, 0.f, 0.f, 0.f, 0.f};
    acc[1] = acc[0];

    // A row r = b*5 + nn -> feats[b, 512+nn, :]
    const float* arow[2];
    bool valid[2];
    #pragma unroll
    for (int ms = 0; ms < 2; ++ms) {
        const int r = ms * 16 + lrow;
        valid[ms] = (r < NOISE_ROWS);
        const int rr = valid[ms] ? r : 0;
        const int bb = rr / NCLUT, nn = rr % NCLUT;
        arow[ms] = feats + (size_t)(bb * FEAT_ROWS + MAXN + nn) * VDIM;
    }

    for (int k = 0; k < VDIM; k += 4) {
        v2f bf;
        {
            const float* p = bptr + (size_t)(k + khalf) * MAXN + lrow;
            bf.x = p[0];
            bf.y = p[MAXN];
        }
        #pragma unroll
        for (int ms = 0; ms < 2; ++ms) {
            v2f af = *(const v2f*)(arow[ms] + k + khalf);
            if (!valid[ms]) { af.x = 0.f; af.y = 0.f; }
            acc[ms] = __builtin_amdgcn_wmma_f32_16x16x4_f32(
                false, af, false, bf, (short)0, acc[ms], false, false);
        }
    }

    const int rofs = (lane >> 4) << 3;
    #pragma unroll
    for (int ms = 0; ms < 2; ++ms)
        #pragma unroll
        for (int r = 0; r < 8; ++r) {
            const int m = ms * 16 + r + rofs;
            if (m < NOISE_ROWS)
                out[(size_t)m * CJ + nBase + lrow] = acc[ms][r];
        }
}

// ---------------------------------------------------------------------------
// new_memory[c,v,j] = l2n_v(0.9*memory + 0.1*visible_kp)
// One block (128 threads = 4 waves, one per v) per (c,j) column.
// ---------------------------------------------------------------------------
__global__ __launch_bounds__(128) void update_memory_kernel(
    const float* __restrict__ feats,
    const unsigned char* __restrict__ visible,
    const int* __restrict__ label,
    const float* __restrict__ memory,
    float* __restrict__ out)
{
    const int cj = blockIdx.x;
    const int c = cj >> 9, j = cj & 511;
    const int v = threadIdx.x;

    int l0 = label[0], l1 = label[1], l2 = label[2], l3 = label[3];
    int cnt = (l0 == c) + (l1 == c) + (l2 == c) + (l3 == c);

    float kp = 0.f;
    if (cnt > 0) {
        const float w = 1.0f / (float)cnt;
        const int lab[4] = {l0, l1, l2, l3};
        #pragma unroll
        for (int b = 0; b < 4; ++b)
            if (lab[b] == c && visible[b * MAXN + j])
                kp += w * feats[(size_t)(b * FEAT_ROWS + j) * VDIM + v];
    }

    const float mem = memory[(size_t)c * (VDIM * MAXN) + (size_t)v * MAXN + j];
    const float val = (cnt > 0) ? (0.9f * mem + 0.1f * kp) : mem;

    float s = val * val;
    #pragma unroll
    for (int off = 16; off > 0; off >>= 1)
        s += __shfl_xor(s, off, 32);
    __shared__ float part[4];
    if ((threadIdx.x & 31) == 0) part[threadIdx.x >> 5] = s;
    __syncthreads();
    const float tot = part[0] + part[1] + part[2] + part[3];
    const float norm = fmaxf(sqrtf(tot), 1e-12f);

    out[(size_t)c * (VDIM * MAXN) + (size_t)v * MAXN + j] = val / norm;
}

// ---------------------------------------------------------------------------
// new_clutter_bank: splice 20 noise columns at new_lru*20, then column l2n.
// One block (128 threads, one per v) per column.
// ---------------------------------------------------------------------------
__global__ __launch_bounds__(128) void update_clutter_kernel(
    const float* __restrict__ feats,
    const float* __restrict__ clutter,
    const int* __restrict__ lru,
    float* __restrict__ out)
{
    const int col = blockIdx.x;
    const int v = threadIdx.x;
    const int new_lru = (lru[0] + 1) % (BANK / 4);   // % 1024
    const int start = new_lru * NOISE_ROWS;          // 20 columns

    float val;
    if (col >= start && col < start + NOISE_ROWS) {
        const int r = col - start;
        const int b = r / NCLUT, nn = r % NCLUT;
        val = feats[(size_t)(b * FEAT_ROWS + MAXN + nn) * VDIM + v];
    } else {
        val = clutter[(size_t)v * CLUT_COLS + col];
    }

    float s = val * val;
    #pragma unroll
    for (int off = 16; off > 0; off >>= 1)
        s += __shfl_xor(s, off, 32);
    __shared__ float part[4];
    if ((threadIdx.x & 31) == 0) part[threadIdx.x >> 5] = s;
    __syncthreads();
    const float tot = part[0] + part[1] + part[2] + part[3];
    const float norm = fmaxf(sqrtf(tot), 1e-12f);

    out[(size_t)v * CLUT_COLS + col] = val / norm;
}

// ---------------------------------------------------------------------------

extern "C" void kernel_launch(void* const* d_in, const int* in_sizes, int n_in,
                              void* d_out, int out_size, void* d_ws, size_t ws_size,
                              hipStream_t stream)
{
    const float*         feats   = (const float*)d_in[0];
    const unsigned char* visible = (const unsigned char*)d_in[1];
    const int*           label   = (const int*)d_in[2];
    const float*         memory  = (const float*)d_in[3];
    const float*         clutter = (const float*)d_in[4];
    const int*           lru     = (const int*)d_in[5];

    float* out  = (float*)d_out;
    float* sim   = out;                                  // 109,051,904
    float* nsim  = out + (size_t)4 * MAXN * SIM_COLS;    // +655,360
    float* nmem  = nsim + (size_t)4 * NCLUT * CJ;        // +4,194,304
    float* nclut = nmem + (size_t)NCLS * VDIM * MAXN;    // +2,621,440

    dim3 gSim(SIM_COLS / 32, 4, 1);                      // 1664 x 4 blocks
    sim_gemm_kernel<<<gSim, 256, 0, stream>>>(feats, memory, clutter, sim);

    noise_gemm_kernel<<<CJ / 16, 32, 0, stream>>>(feats, memory, nsim);

    update_memory_kernel<<<NCLS * MAXN, 128, 0, stream>>>(feats, visible, label,
                                                          memory, nmem);

    update_clutter_kernel<<<CLUT_COLS, 128, 0, stream>>>(feats, clutter, lru, nclut);
}